// LogHarmonicLowering_28363964023075
// MI455X (gfx1250) — compile-verified
//
#include <hip/hip_runtime.h>
#include <math.h>

// ---------------------------------------------------------------------------
// LogHarmonicLowering on MI455X (gfx1250).
// out[b, k*C+c, f, t] = w0*x[b,c,i0,t] + w1*x[b,c,i0+1,t], i0 = floor(f - shift[k]),
// zero outside [0,F). Pure HBM-bandwidth problem: 128MiB in (fits 192MB L2),
// 640MiB out -> ~33us floor @ 23.3TB/s. Design minimizes instructions/byte:
//  * TDM (tensor_load_to_lds): one DMA per 17x256 shift window into LDS.
//    D# tensor_dim1 = max(F - s0, 0) => hardware zero-fill for OOB rows,
//    no per-element load instructions, TENSORcnt + s_wait_tensorcnt sync.
//  * 80-entry (k,row) weight/offset table precomputed once per block in LDS:
//    each row task reads one b128 instead of redoing floor/frac/cndmask VALU.
//  * Compute: ds_load_b128 row reads (512B/instr), packed v_pk_fma_f32 math,
//    b128 non-temporal stores so the output stream never evicts the
//    L2-resident input.
// ---------------------------------------------------------------------------

typedef float f4  __attribute__((ext_vector_type(4)));
typedef unsigned v4u __attribute__((ext_vector_type(4)));
typedef int  v4i __attribute__((ext_vector_type(4)));
typedef int  v8i __attribute__((ext_vector_type(8)));

#define BDIM 4
#define CDIM 16
#define FDIM 1024
#define TDIM 512
#define KDIM 5
#define TS   256               // time columns per workgroup
#define SLAB_ROWS 17           // 16 output rows need rows s0..s0+16
#define SLAB_K (SLAB_ROWS * TS)     // floats per shift window slab

struct ShiftArgs {
    float shift[KDIM];  // shifts[k] as float32 (non-positive), matches numpy cast
    int   d[KDIM];      // floor(-shift) = integer source row offset (>= 0)
};

__global__ __launch_bounds__(256) void log_harmonic_tdm_kernel(
    const float* __restrict__ x, float* __restrict__ out, ShiftArgs sa)
{
    __shared__ __align__(16) float slab[KDIM * SLAB_K];   // 5*17*256*4B = 85KB
    __shared__ __align__(16) f4 tab[KDIM * 16];           // per-(k,row) weights/offsets

    const int tid = threadIdx.x;
    unsigned bid = blockIdx.x;
    const int th = bid & 1;   bid >>= 1;   // time half: cols [th*256, th*256+256)
    const int ft = bid & 63;  bid >>= 6;   // 64 freq tiles of 16 rows
    const int bc = bid;                    // b*C + c, 0..63
    const int f0 = ft * 16;
    const int tcol0 = th * TS;

#if __has_builtin(__builtin_amdgcn_tensor_load_to_lds)
    // ---- TDM stage: wave 0 DMAs the 5 shift windows into LDS ----
    if (tid < 32) {
        const unsigned lds0 = (unsigned)(size_t)(void*)&slab[0];  // low 32b = LDS addr
        const unsigned long long xb = (unsigned long long)(size_t)x;
#pragma unroll
        for (int k = 0; k < KDIM; ++k) {
            const int s0 = f0 + sa.d[k];
            const int nv = FDIM - s0;                       // valid rows from s0
            const unsigned tdim1 = nv > 0 ? (unsigned)nv : 0u;
            const unsigned long long ga =
                xb + 4ull * ((unsigned long long)(bc * FDIM + s0) * TDIM + tcol0);
            v4u g0; v8i g1;
            v4i gz4 = {0, 0, 0, 0};
            v8i gz8 = {0, 0, 0, 0, 0, 0, 0, 0};
            g0[0] = 1u;                                     // count=1 (valid user D#)
            g0[1] = lds0 + (unsigned)k * (SLAB_K * 4u);     // lds_addr (bytes)
            g0[2] = (unsigned)ga;                           // global_addr[31:0]
            g0[3] = (unsigned)((ga >> 32) & 0x01FFFFFFull)  // global_addr[56:32]
                    | (2u << 30);                           // type=2 (image)
            g1[0] = (int)(2u << 16);                        // data_size=2 -> 4 bytes
            g1[1] = (int)((unsigned)TS << 16);              // tensor_dim0[15:0]=256
            g1[2] = (int)(tdim1 << 16);                     // tensor_dim1: OOB rows read 0
            g1[3] = (int)((unsigned)TS << 16);              // tile_dim0=256
            g1[4] = SLAB_ROWS;                              // tile_dim1=17 rows
            g1[5] = TDIM;                                   // tensor_dim0_stride=512
            g1[6] = 0;
            g1[7] = 0;
            __builtin_amdgcn_tensor_load_to_lds(g0, g1, gz4, gz4, gz8, 0);
        }
        __builtin_amdgcn_s_wait_tensorcnt(0);
    }
#else
    // Fallback staging (portable): cooperative loads with manual zero-fill.
    for (int idx = tid; idx < KDIM * SLAB_K; idx += 256) {
        const int k  = idx / SLAB_K;
        const int rm = idx - k * SLAB_K;
        const int rr = rm >> 8;            // row in window
        const int c  = rm & (TS - 1);
        const int row = f0 + sa.d[k] + rr;
        slab[idx] = (row < FDIM) ? x[((size_t)bc * FDIM + row) * TDIM + tcol0 + c]
                                 : 0.0f;
    }
#endif

    // ---- Per-(k,row) weight/offset table: exact f32 replication, once/block ----
    if (tid < KDIM * 16) {
        const int k = tid >> 4;
        const int r = tid & 15;
        const int   fr  = f0 + r;
        const float pos = (float)fr - sa.shift[k];
        const float flo = floorf(pos);
        const float w   = pos - flo;
        const int   i0  = (int)flo;
        const int   s0  = f0 + sa.d[k];
        int j = i0 - s0;                       // == r in practice; clamp for safety
        j = j < 0 ? 0 : (j > 15 ? 15 : j);
        const float w0v = ((unsigned)i0       < (unsigned)FDIM) ? (1.0f - w) : 0.0f;
        const float w1v = ((unsigned)(i0 + 1) < (unsigned)FDIM) ? w          : 0.0f;
        const int srow = k * SLAB_K + j * TS;                    // slab row offset
        const int orow = k * (CDIM * FDIM * TDIM) + r * TDIM;    // out row offset
        f4 e;
        e.x = w0v;
        e.y = w1v;
        e.z = __int_as_float(srow);
        e.w = __int_as_float(orow);
        tab[tid] = e;
    }
    __syncthreads();

    // ---- Compute: 80 (k,row) tasks, 10 per wave ----
    const int wave = tid >> 5;
    const int lane = tid & 31;
    const int b_   = bc >> 4;       // bc / C
    const int cch  = bc & 15;       // bc % C
    float* __restrict__ obase =
        out + ((size_t)(b_ * (KDIM * CDIM) + cch) * FDIM + f0) * TDIM + tcol0;

#pragma unroll
    for (int t = 0; t < 10; ++t) {
        const int task = wave * 10 + t;
        const f4 e = tab[task];                          // uniform ds_load_b128
        const float w0v = e.x;
        const float w1v = e.y;
        const float* __restrict__ row0 = &slab[__float_as_int(e.z)];
        const float* __restrict__ row1 = row0 + TS;
        float* __restrict__ op = obase + __float_as_int(e.w);

        const f4 w0s = {w0v, w0v, w0v, w0v};
        const f4 w1s = {w1v, w1v, w1v, w1v};
#pragma unroll
        for (int it = 0; it < 2; ++it) {
            const int c = lane * 4 + it * 128;
            const f4 g0 = *(const f4*)(row0 + c);                 // ds_load_b128
            const f4 g1 = *(const f4*)(row1 + c);                 // ds_load_b128
            const f4 o  = __builtin_elementwise_fma(g0, w0s, g1 * w1s); // v_pk_fma_f32
            __builtin_nontemporal_store(o, (f4*)(op + c));        // b128 NT store
        }
    }
}

extern "C" void kernel_launch(void* const* d_in, const int* in_sizes, int n_in,
                              void* d_out, int out_size, void* d_ws, size_t ws_size,
                              hipStream_t stream) {
    (void)in_sizes; (void)n_in; (void)d_ws; (void)ws_size; (void)out_size;
    const float* x = (const float*)d_in[0];
    float* out = (float*)d_out;

    // Faithful port of make_log_shift (double precision, cast to f32 like numpy).
    ShiftArgs sa;
    const double anchor_ls = 200.0 * log(0.001 * 1.0);
    for (int n = 1; n <= KDIM; ++n) {
        double ls = 200.0 * log(0.001 * (double)n) - anchor_ls;  // = 200*ln(n) >= 0
        sa.shift[n - 1] = (float)(-ls);
        sa.d[n - 1]     = (int)floor(ls);
    }

    const int blocks = (BDIM * CDIM) * (FDIM / 16) * (TDIM / TS);   // 64*64*2 = 8192
    log_harmonic_tdm_kernel<<<blocks, 256, 0, stream>>>(x, out, sa);
}